// Decoder_65335042506781
// MI455X (gfx1250) — compile-verified
//
#include <hip/hip_runtime.h>
#include <math.h>

#if defined(__HIP_DEVICE_COMPILE__) && !__has_builtin(__builtin_amdgcn_wmma_f32_16x16x4_f32)
#error "wmma f32 16x16x4 builtin not available on this toolchain (device pass)"
#endif

typedef __attribute__((ext_vector_type(2))) float v2f;
typedef __attribute__((ext_vector_type(8))) float v8f;

// ---------------------------------------------------------------- helpers
__device__ __forceinline__ v8f wmma4(v2f a, v2f b, v8f c) {
  return __builtin_amdgcn_wmma_f32_16x16x4_f32(false, a, false, b, (short)0, c, false, false);
}
__device__ __forceinline__ float wsum(float v) {
#pragma unroll
  for (int m = 16; m >= 1; m >>= 1) v += __shfl_xor(v, m, 32);
  return v;
}
__device__ __forceinline__ float wmaxr(float v) {
#pragma unroll
  for (int m = 16; m >= 1; m >>= 1) v = fmaxf(v, __shfl_xor(v, m, 32));
  return v;
}
__device__ __forceinline__ float gelu_f(float x) {
  float x3 = x * x * x;
  return 0.5f * x * (1.f + tanhf(0.7978845608028654f * (x + 0.044715f * x3)));
}
// A fragment: row-major A[M][lda], lane = M (0..15), K pair split across half-waves
__device__ __forceinline__ v2f afrag(const float* A, int lda, int kk) {
  int l = threadIdx.x & 31;
  int m = l & 15;
  int kh = kk + ((l >> 4) << 1);
  v2f r; r.x = A[m * lda + kh]; r.y = A[m * lda + kh + 1];
  return r;
}
// B fragment: row-major B[K][ldb], lane = N (0..15), K pair split across half-waves
__device__ __forceinline__ v2f bfrag(const float* B, int ldb, int kk) {
  int l = threadIdx.x & 31;
  int n = l & 15;
  int kh = kk + ((l >> 4) << 1);
  v2f r; r.x = B[kh * ldb + n]; r.y = B[(kh + 1) * ldb + n];
  return r;
}

// ------------------------------------------------ generic fp32 WMMA GEMM
// C[M,N] = (accum ? C : 0) + A[M,K] @ W[K,N] + (add_bias ? bias[N] : 0)
__global__ void __launch_bounds__(256) gemm_kernel(const float* __restrict__ A,
                                                   const float* __restrict__ W,
                                                   const float* __restrict__ bias,
                                                   float* __restrict__ C,
                                                   int M, int N, int K,
                                                   int add_bias, int accum) {
  __shared__ float As[32][36];
  __shared__ float Ws[32][68];
  int tid = threadIdx.x, lane = tid & 31, w = tid >> 5;
  int wm = w >> 2, wn = w & 3;                 // 2x4 wave tiles
  int m0 = blockIdx.x * 32, n0 = blockIdx.y * 64;
  v8f acc = {0.f, 0.f, 0.f, 0.f, 0.f, 0.f, 0.f, 0.f};
  for (int kb = 0; kb < K; kb += 32) {
    for (int t = tid; t < 32 * 32; t += 256) {
      int r = t >> 5, c = t & 31;
      As[r][c] = A[(size_t)(m0 + r) * K + kb + c];
    }
    for (int t = tid; t < 32 * 64; t += 256) {
      int r = t >> 6, c = t & 63;
      Ws[r][c] = W[(size_t)(kb + r) * N + n0 + c];
    }
    __syncthreads();
    const float* Ap = &As[wm * 16][0];
    const float* Bp = &Ws[0][wn * 16];
#pragma unroll
    for (int kk = 0; kk < 32; kk += 4)
      acc = wmma4(afrag(Ap, 36, kk), bfrag(Bp, 68, kk), acc);
    __syncthreads();
  }
  int n = n0 + wn * 16 + (lane & 15);
  float bv = add_bias ? bias[n] : 0.f;
#pragma unroll
  for (int e = 0; e < 8; e++) {
    int m = m0 + wm * 16 + e + ((lane >> 4) << 3);
    float val = acc[e] + bv;
    if (accum) val += C[(size_t)m * N + n];
    C[(size_t)m * N + n] = val;
  }
}

// ------------------------------------------------ LayerNorm over 256 cols
__global__ void __launch_bounds__(256) ln256_kernel(const float* __restrict__ X,
                                                    float* __restrict__ Y,
                                                    const float* __restrict__ sc,
                                                    const float* __restrict__ of) {
  int row = blockIdx.x, t = threadIdx.x, lane = t & 31, w = t >> 5;
  __shared__ float r1[8], r2[8];
  float x = X[(size_t)row * 256 + t];
  float s1 = wsum(x), s2 = wsum(x * x);
  if (!lane) { r1[w] = s1; r2[w] = s2; }
  __syncthreads();
  float m = 0.f, v = 0.f;
  for (int u = 0; u < 8; u++) { m += r1[u]; v += r2[u]; }
  m *= (1.f / 256.f);
  v = v * (1.f / 256.f) - m * m;
  float y = (x - m) * rsqrtf(v + 1e-5f);
  if (sc) y = y * sc[t] + of[t];
  Y[(size_t)row * 256 + t] = y;
}

// ------------------------------------------------ per-head LN for q (scaled) and k
__global__ void __launch_bounds__(256) headln_kernel(float* __restrict__ q,
                                                     float* __restrict__ k) {
  int idx = blockIdx.x * 8 + (threadIdx.x >> 5);
  int lane = threadIdx.x & 31;
  const int NH = 512 * 8;
  float* base;
  float scale;
  int r;
  if (idx < NH) { base = q; r = idx; scale = rsqrtf(32.f + 1e-6f); }
  else { base = k; r = idx - NH; scale = 1.f; }
  int n = r >> 3, h = r & 7;
  float* p = base + (size_t)n * 256 + h * 32;
  float x = p[lane];
  float s1 = wsum(x), s2 = wsum(x * x);
  float m = s1 * (1.f / 32.f), v = s2 * (1.f / 32.f) - m * m;
  p[lane] = (x - m) * rsqrtf(v + 1e-5f) * scale;
}

// ------------------------------------------------ pair prep: lnp -> bias(Wb) + pairv(Wp)
// logits[h][i][j] = LN(pair[i,j]) @ Wb[:,h] ; pairv[i][j][s] = LN(pair[i,j]) @ Wp[:,s]
__global__ void __launch_bounds__(256) pairprep_kernel(const float* __restrict__ pair,
                                                       const float* __restrict__ Wb,
                                                       const float* __restrict__ Wp,
                                                       float* __restrict__ logits,
                                                       float* __restrict__ pairv) {
  __shared__ float lnps[64][66];
  __shared__ float Wc[64][48];
  int i = blockIdx.x, j0 = blockIdx.y * 64;
  int tid = threadIdx.x, lane = tid & 31, w = tid >> 5;
  for (int t = tid; t < 64 * 48; t += 256) {
    int r = t / 48, c = t % 48;
    float val = 0.f;
    if (c < 8) val = Wb[r * 8 + c];
    else if (c < 40) val = Wp[r * 32 + (c - 8)];
    Wc[r][c] = val;
  }
  for (int rr = 0; rr < 8; rr++) {
    int r = w * 8 + rr, j = j0 + r;
    const float* p = pair + ((size_t)i * 512 + j) * 64;
    float x0 = p[2 * lane], x1 = p[2 * lane + 1];
    float s1 = wsum(x0 + x1), s2 = wsum(x0 * x0 + x1 * x1);
    float m = s1 * (1.f / 64.f), v = s2 * (1.f / 64.f) - m * m, rs = rsqrtf(v + 1e-5f);
    lnps[r][2 * lane] = (x0 - m) * rs;
    lnps[r][2 * lane + 1] = (x1 - m) * rs;
  }
  __syncthreads();
  for (int t = w; t < 12; t += 8) {
    int mt = t & 3, nt = t >> 2;  // 4 m-tiles x 3 n-tiles
    v8f acc = {0.f, 0.f, 0.f, 0.f, 0.f, 0.f, 0.f, 0.f};
    const float* Ap = &lnps[mt * 16][0];
    const float* Bp = &Wc[0][nt * 16];
#pragma unroll
    for (int kk = 0; kk < 64; kk += 4)
      acc = wmma4(afrag(Ap, 66, kk), bfrag(Bp, 48, kk), acc);
    int nn = nt * 16 + (lane & 15);
#pragma unroll
    for (int e = 0; e < 8; e++) {
      int r = mt * 16 + e + ((lane >> 4) << 3);
      int j = j0 + r;
      if (nn < 8) logits[((size_t)nn * 512 + i) * 512 + j] = acc[e];
      else if (nn < 40) pairv[((size_t)i * 512 + j) * 32 + (nn - 8)] = acc[e];
    }
  }
}

// ------------------------------------------------ scores: logits[h] += q_h @ k_h^T
__global__ void __launch_bounds__(128) scores_kernel(const float* __restrict__ q,
                                                     const float* __restrict__ k,
                                                     float* __restrict__ logits) {
  int it = blockIdx.x, jb = blockIdx.y, h = blockIdx.z;
  int lane = threadIdx.x & 31, w = threadIdx.x >> 5;
  int i0 = it * 16, j0 = jb * 64 + w * 16;
  float* Lp = logits + ((size_t)h * 512 + i0) * 512 + j0;
  v8f acc;
#pragma unroll
  for (int e = 0; e < 8; e++) {
    int m = e + ((lane >> 4) << 3);
    acc[e] = Lp[(size_t)m * 512 + (lane & 15)];
  }
  const float* Ap = q + (size_t)i0 * 256 + h * 32;
  const float* Bp = k + (size_t)j0 * 256 + h * 32;
  int n = lane & 15;
#pragma unroll
  for (int kk = 0; kk < 32; kk += 4) {
    v2f a = afrag(Ap, 256, kk);
    int kh = kk + ((lane >> 4) << 1);
    v2f b; b.x = Bp[(size_t)n * 256 + kh]; b.y = Bp[(size_t)n * 256 + kh + 1];
    acc = wmma4(a, b, acc);
  }
#pragma unroll
  for (int e = 0; e < 8; e++) {
    int m = e + ((lane >> 4) << 3);
    Lp[(size_t)m * 512 + (lane & 15)] = acc[e];
  }
}

// ------------------------------------------------ masked softmax over j
__global__ void __launch_bounds__(256) softmax_kernel(float* __restrict__ logits,
                                                      const int* __restrict__ mask) {
  int i = blockIdx.x, h = blockIdx.y, t = threadIdx.x;
  int lane = t & 31, w = t >> 5;
  float* row = logits + ((size_t)h * 512 + i) * 512;
  __shared__ float red[8];
  bool mi = mask[i] != 0;
  bool p0 = mi && (mask[t] != 0);
  bool p1 = mi && (mask[t + 256] != 0);
  float x0 = p0 ? row[t] : -1e9f;
  float x1 = p1 ? row[t + 256] : -1e9f;
  float mx = wmaxr(fmaxf(x0, x1));
  if (!lane) red[w] = mx;
  __syncthreads();
  float gmx = -3.4e38f;
  for (int u = 0; u < 8; u++) gmx = fmaxf(gmx, red[u]);
  float e0 = expf(x0 - gmx), e1 = expf(x1 - gmx);
  float s = wsum(e0 + e1);
  __syncthreads();
  if (!lane) red[w] = s;
  __syncthreads();
  float gs = 0.f;
  for (int u = 0; u < 8; u++) gs += red[u];
  float inv = 1.f / gs;
  row[t] = p0 ? e0 * inv : 0.f;
  row[t + 256] = p1 ? e1 * inv : 0.f;
}

// ------------------------------------------------ out = attn @ v  (attcat cols 0..255)
__global__ void __launch_bounds__(256) attnv_kernel(const float* __restrict__ attn,
                                                    const float* __restrict__ v,
                                                    float* __restrict__ attcat) {
  int tile = blockIdx.x * 8 + (threadIdx.x >> 5);   // 512 tiles total
  int lane = threadIdx.x & 31;
  int h = tile >> 6;
  int rem = tile & 63;
  int it = rem >> 1, st = rem & 1;
  int i0 = it * 16, s0 = st * 16;
  const float* Ap = attn + ((size_t)h * 512 + i0) * 512;
  v8f acc = {0.f, 0.f, 0.f, 0.f, 0.f, 0.f, 0.f, 0.f};
  int n = lane & 15;
  for (int kk = 0; kk < 512; kk += 4) {
    v2f a = afrag(Ap, 512, kk);
    int kh = kk + ((lane >> 4) << 1);
    v2f b;
    b.x = v[(size_t)kh * 256 + h * 32 + s0 + n];
    b.y = v[(size_t)(kh + 1) * 256 + h * 32 + s0 + n];
    acc = wmma4(a, b, acc);
  }
#pragma unroll
  for (int e = 0; e < 8; e++) {
    int m = i0 + e + ((lane >> 4) << 3);
    attcat[(size_t)m * 512 + h * 32 + s0 + (lane & 15)] = acc[e];
  }
}

// ------------------------------------------------ out_pair (attcat cols 256..511)
__global__ void __launch_bounds__(256) attnpair_kernel(const float* __restrict__ attn,
                                                       const float* __restrict__ pairv,
                                                       float* __restrict__ attcat) {
  int i = blockIdx.x;
  int h = threadIdx.x >> 5, s = threadIdx.x & 31;
  const float* arow = attn + ((size_t)h * 512 + i) * 512;
  const float* pv = pairv + (size_t)i * 512 * 32 + s;
  float acc = 0.f;
#pragma unroll 4
  for (int j = 0; j < 512; j++) acc += arow[j] * pv[(size_t)j * 32];
  attcat[(size_t)i * 512 + 256 + h * 32 + s] = acc;
}

// ------------------------------------------------ h = gelu(g) * d
__global__ void gelugate_kernel(const float* __restrict__ g, const float* __restrict__ d,
                                float* __restrict__ h, int n) {
  int idx = blockIdx.x * 256 + threadIdx.x;
  if (idx < n) h[idx] = gelu_f(g[idx]) * d[idx];
}

// ------------------------------------------------ LocalToPair: pg/pv via WMMA, reduce left/right
__global__ void __launch_bounds__(256) l2p_kernel(const float* __restrict__ pair,
                                                  const float* __restrict__ Pg,
                                                  const float* __restrict__ Pv,
                                                  const float* __restrict__ gl,
                                                  const float* __restrict__ vl,
                                                  const float* __restrict__ gr,
                                                  const float* __restrict__ vr,
                                                  const int* __restrict__ mask,
                                                  float* __restrict__ left,
                                                  float* __restrict__ right) {
  __shared__ float lnps[128][66];
  __shared__ float rightp[8][132];
  int i0 = blockIdx.x * 16, j0 = blockIdx.y * 8;
  int tid = threadIdx.x, lane = tid & 31, w = tid >> 5;
  for (int t = tid; t < 8 * 132; t += 256) (&rightp[0][0])[t] = 0.f;
  for (int rr = 0; rr < 16; rr++) {
    int r = w * 16 + rr;
    int il = r >> 3, jl = r & 7;
    const float* p = pair + (((size_t)(i0 + il)) * 512 + (j0 + jl)) * 64;
    float x0 = p[2 * lane], x1 = p[2 * lane + 1];
    float s1 = wsum(x0 + x1), s2 = wsum(x0 * x0 + x1 * x1);
    float m = s1 * (1.f / 64.f), vv = s2 * (1.f / 64.f) - m * m, rs = rsqrtf(vv + 1e-5f);
    lnps[r][2 * lane] = (x0 - m) * rs;
    lnps[r][2 * lane + 1] = (x1 - m) * rs;
  }
  __syncthreads();
  int nn = w * 16 + (lane & 15);   // each wave owns a private 16-column band
  int hi = (lane >> 4);
  for (int mt = 0; mt < 8; mt++) {
    v8f apg = {0.f, 0.f, 0.f, 0.f, 0.f, 0.f, 0.f, 0.f};
    v8f apv = {0.f, 0.f, 0.f, 0.f, 0.f, 0.f, 0.f, 0.f};
    const float* Ap = &lnps[mt * 16][0];
#pragma unroll
    for (int kk = 0; kk < 64; kk += 4) {
      v2f a = afrag(Ap, 66, kk);
      int kh = kk + ((lane >> 4) << 1);
      v2f b; b.x = Pg[kh * 128 + nn]; b.y = Pg[(kh + 1) * 128 + nn];
      apg = wmma4(a, b, apg);
    }
#pragma unroll
    for (int kk = 0; kk < 64; kk += 4) {
      v2f a = afrag(Ap, 66, kk);
      int kh = kk + ((lane >> 4) << 1);
      v2f b; b.x = Pv[kh * 128 + nn]; b.y = Pv[(kh + 1) * 128 + nn];
      apv = wmma4(a, b, apv);
    }
    int il = 2 * mt + hi;
    int i = i0 + il;
    bool maski = mask[i] != 0;
    float gli = gl[(size_t)i * 128 + nn];
    float vri = vr[(size_t)i * 128 + nn];
    float leftsum = 0.f;
#pragma unroll
    for (int e = 0; e < 8; e++) {
      int j = j0 + e;
      bool pm = maski && (mask[j] != 0);
      if (pm) {
        float pgv = apg[e], pvv = apv[e];
        float vlj = vl[(size_t)j * 128 + nn];
        float grj = gr[(size_t)j * 128 + nn];
        leftsum += gelu_f(gli + pgv) * (vlj + pvv);
        atomicAdd(&rightp[e][nn], gelu_f(grj + pgv) * (vri + pvv));
      }
    }
    atomicAdd(&left[(size_t)i * 128 + nn], leftsum);
  }
  __syncthreads();
  for (int t = tid; t < 8 * 128; t += 256) {
    int jl = t >> 7, c = t & 127;
    atomicAdd(&right[(size_t)(j0 + jl) * 128 + c], rightp[jl][c]);
  }
}

// ------------------------------------------------ fused: pair += concat(lnp, LN(left+right)) @ Wl2p,
// then Update(pair): x=LN(*, s, b); pair += (gelu(x@U2g)*(x@U2d)) @ U2o
__global__ void __launch_bounds__(256) pairout_kernel(float* __restrict__ pair,
                                                      const float* __restrict__ left,
                                                      const float* __restrict__ right,
                                                      const float* __restrict__ Wl2p,
                                                      const float* __restrict__ l2s,
                                                      const float* __restrict__ l2b,
                                                      const float* __restrict__ U2g,
                                                      const float* __restrict__ U2d,
                                                      const float* __restrict__ U2o) {
  __shared__ float xbuf[32][200];
  __shared__ float obuf[32][68];
  int i = blockIdx.x, j0 = blockIdx.y * 32;
  int tid = threadIdx.x, lane = tid & 31, w = tid >> 5;
  // phase 1: x192 = [ LN(pair row) | LN(left[i]+right[j]) ]
  for (int rr = 0; rr < 4; rr++) {
    int r = w * 4 + rr, j = j0 + r;
    const float* p = pair + ((size_t)i * 512 + j) * 64;
    float x0 = p[2 * lane], x1 = p[2 * lane + 1];
    float s1 = wsum(x0 + x1), s2 = wsum(x0 * x0 + x1 * x1);
    float m = s1 * (1.f / 64.f), vv = s2 * (1.f / 64.f) - m * m, rs = rsqrtf(vv + 1e-5f);
    xbuf[r][2 * lane] = (x0 - m) * rs;
    xbuf[r][2 * lane + 1] = (x1 - m) * rs;
    float t0 = left[(size_t)i * 128 + lane] + right[(size_t)j * 128 + lane];
    float t1 = left[(size_t)i * 128 + 32 + lane] + right[(size_t)j * 128 + 32 + lane];
    float t2 = left[(size_t)i * 128 + 64 + lane] + right[(size_t)j * 128 + 64 + lane];
    float t3 = left[(size_t)i * 128 + 96 + lane] + right[(size_t)j * 128 + 96 + lane];
    float u1 = wsum(t0 + t1 + t2 + t3);
    float u2 = wsum(t0 * t0 + t1 * t1 + t2 * t2 + t3 * t3);
    float mm = u1 * (1.f / 128.f), v2 = u2 * (1.f / 128.f) - mm * mm, r2 = rsqrtf(v2 + 1e-5f);
    xbuf[r][64 + lane] = (t0 - mm) * r2;
    xbuf[r][96 + lane] = (t1 - mm) * r2;
    xbuf[r][128 + lane] = (t2 - mm) * r2;
    xbuf[r][160 + lane] = (t3 - mm) * r2;
  }
  __syncthreads();
  // phase 2: pair1 = pair + x192 @ Wl2p   (2x4 wave tiles)
  {
    int mt = w >> 2, nt = w & 3;
    const float* Ap = &xbuf[mt * 16][0];
    int nn = nt * 16 + (lane & 15);
    v8f acc;
#pragma unroll
    for (int e = 0; e < 8; e++) {
      int r = mt * 16 + e + ((lane >> 4) << 3);
      acc[e] = pair[((size_t)i * 512 + j0 + r) * 64 + nn];
    }
    for (int kk = 0; kk < 192; kk += 4) {
      v2f a = afrag(Ap, 200, kk);
      int kh = kk + ((lane >> 4) << 1);
      v2f b; b.x = Wl2p[kh * 64 + nn]; b.y = Wl2p[(kh + 1) * 64 + nn];
      acc = wmma4(a, b, acc);
    }
#pragma unroll
    for (int e = 0; e < 8; e++) {
      int r = mt * 16 + e + ((lane >> 4) << 3);
      obuf[r][nn] = acc[e];
    }
  }
  __syncthreads();
  // phase 3: x2 = LN(pair1)*s + b  -> xbuf[:,0:64]
  for (int rr = 0; rr < 4; rr++) {
    int r = w * 4 + rr;
    float x0 = obuf[r][2 * lane], x1 = obuf[r][2 * lane + 1];
    float s1 = wsum(x0 + x1), s2 = wsum(x0 * x0 + x1 * x1);
    float m = s1 * (1.f / 64.f), vv = s2 * (1.f / 64.f) - m * m, rs = rsqrtf(vv + 1e-5f);
    xbuf[r][2 * lane] = (x0 - m) * rs * l2s[2 * lane] + l2b[2 * lane];
    xbuf[r][2 * lane + 1] = (x1 - m) * rs * l2s[2 * lane + 1] + l2b[2 * lane + 1];
  }
  __syncthreads();
  // phase 4: h = gelu(x2@U2g) * (x2@U2d)  -> xbuf[:,64:192]
  for (int t = w; t < 16; t += 8) {
    int mt = t >> 3, nt = t & 7;
    const float* Ap = &xbuf[mt * 16][0];
    int nn = nt * 16 + (lane & 15);
    v8f ag = {0.f, 0.f, 0.f, 0.f, 0.f, 0.f, 0.f, 0.f};
    v8f ad = {0.f, 0.f, 0.f, 0.f, 0.f, 0.f, 0.f, 0.f};
#pragma unroll
    for (int kk = 0; kk < 64; kk += 4) {
      v2f a = afrag(Ap, 200, kk);
      int kh = kk + ((lane >> 4) << 1);
      v2f b; b.x = U2g[kh * 128 + nn]; b.y = U2g[(kh + 1) * 128 + nn];
      ag = wmma4(a, b, ag);
    }
#pragma unroll
    for (int kk = 0; kk < 64; kk += 4) {
      v2f a = afrag(Ap, 200, kk);
      int kh = kk + ((lane >> 4) << 1);
      v2f b; b.x = U2d[kh * 128 + nn]; b.y = U2d[(kh + 1) * 128 + nn];
      ad = wmma4(a, b, ad);
    }
#pragma unroll
    for (int e = 0; e < 8; e++) {
      int r = mt * 16 + e + ((lane >> 4) << 3);
      xbuf[r][64 + nn] = gelu_f(ag[e]) * ad[e];
    }
  }
  __syncthreads();
  // phase 5: pair = pair1 + h @ U2o
  {
    int mt = w >> 2, nt = w & 3;
    int nn = nt * 16 + (lane & 15);
    const float* Ap = &xbuf[mt * 16][64];
    v8f acc;
#pragma unroll
    for (int e = 0; e < 8; e++) {
      int r = mt * 16 + e + ((lane >> 4) << 3);
      acc[e] = obuf[r][nn];
    }
    for (int kk = 0; kk < 128; kk += 4) {
      v2f a = afrag(Ap, 200, kk);
      int kh = kk + ((lane >> 4) << 1);
      v2f b; b.x = U2o[kh * 64 + nn]; b.y = U2o[(kh + 1) * 64 + nn];
      acc = wmma4(a, b, acc);
    }
#pragma unroll
    for (int e = 0; e < 8; e++) {
      int r = mt * 16 + e + ((lane >> 4) << 3);
      pair[((size_t)i * 512 + j0 + r) * 64 + nn] = acc[e];
    }
  }
}

// ------------------------------------------------ final LN over P=64 rows (in place)
__global__ void __launch_bounds__(256) lnpair_kernel(float* __restrict__ Pm,
                                                     const float* __restrict__ sc,
                                                     const float* __restrict__ of) {
  int w = threadIdx.x >> 5, lane = threadIdx.x & 31;
  size_t row = (size_t)blockIdx.x * 8 + w;
  float* p = Pm + row * 64;
  float x0 = p[2 * lane], x1 = p[2 * lane + 1];
  float s1 = wsum(x0 + x1), s2 = wsum(x0 * x0 + x1 * x1);
  float m = s1 * (1.f / 64.f), v = s2 * (1.f / 64.f) - m * m, r = rsqrtf(v + 1e-5f);
  p[2 * lane] = (x0 - m) * r * sc[2 * lane] + of[2 * lane];
  p[2 * lane + 1] = (x1 - m) * r * sc[2 * lane + 1] + of[2 * lane + 1];
}

// ================================================================ host
extern "C" void kernel_launch(void* const* d_in, const int* in_sizes, int n_in,
                              void* d_out, int out_size, void* d_ws, size_t ws_size,
                              hipStream_t stream) {
  (void)in_sizes; (void)n_in; (void)out_size; (void)ws_size;
  auto F = [&](int i) { return (const float*)d_in[i]; };
  const int* mask = (const int*)d_in[2];

  float* out = (float*)d_out;
  float* local = out;                 // [512,256]
  float* pair = out + 512 * 256;      // [512,512,64]
  (void)hipMemcpyAsync(local, d_in[0], (size_t)512 * 256 * sizeof(float),
                       hipMemcpyDeviceToDevice, stream);
  (void)hipMemcpyAsync(pair, d_in[1], (size_t)512 * 512 * 64 * sizeof(float),
                       hipMemcpyDeviceToDevice, stream);

  float* ws = (float*)d_ws;
  float* lnl = ws;       ws += 131072;
  float* qb = ws;        ws += 131072;
  float* kb = ws;        ws += 131072;
  float* vb = ws;        ws += 131072;
  float* logits = ws;    ws += 2097152;   // [H][N][N]
  float* pairv = ws;     ws += 8388608;   // [N][N][S]
  float* attcat = ws;    ws += 262144;    // [N][512]
  float* xb = ws;        ws += 131072;
  float* gbuf = ws;      ws += 262144;
  float* dbuf = ws;      ws += 262144;
  float* hbuf = ws;      ws += 262144;
  float* gl = ws;        ws += 65536;
  float* vl = ws;        ws += 65536;
  float* gr = ws;        ws += 65536;
  float* vrb = ws;       ws += 65536;
  float* left = ws;      ws += 65536;
  float* right = ws;     ws += 65536;

  for (int l = 0; l < 2; l++) {
    const float* Wq = F(3) + (size_t)l * 65536;   const float* bq = F(4) + l * 256;
    const float* Wk = F(5) + (size_t)l * 65536;   const float* bk = F(6) + l * 256;
    const float* Wv = F(7) + (size_t)l * 65536;   const float* bv = F(8) + l * 256;
    const float* Wb = F(9) + l * 512;
    const float* Wp = F(10) + l * 2048;
    const float* Wo = F(11) + (size_t)l * 131072;
    const float* l1s = F(12) + l * 256;           const float* l1b = F(13) + l * 256;
    const float* U1g = F(14) + (size_t)l * 131072;
    const float* U1d = F(15) + (size_t)l * 131072;
    const float* U1o = F(16) + (size_t)l * 131072;
    const float* Pg = F(17) + l * 8192;           const float* Pv = F(18) + l * 8192;
    const float* Lgl = F(19) + (size_t)l * 32768; const float* Lvl = F(20) + (size_t)l * 32768;
    const float* Lgr = F(21) + (size_t)l * 32768; const float* Lvr = F(22) + (size_t)l * 32768;
    const float* Wl2p = F(23) + l * 12288;
    const float* l2s = F(24) + l * 64;            const float* l2b = F(25) + l * 64;
    const float* U2g = F(26) + l * 8192;          const float* U2d = F(27) + l * 8192;
    const float* U2o = F(28) + l * 8192;

    // --- DecoderAttention ---
    ln256_kernel<<<512, 256, 0, stream>>>(local, lnl, nullptr, nullptr);
    gemm_kernel<<<dim3(16, 4), 256, 0, stream>>>(lnl, Wq, bq, qb, 512, 256, 256, 1, 0);
    gemm_kernel<<<dim3(16, 4), 256, 0, stream>>>(lnl, Wk, bk, kb, 512, 256, 256, 1, 0);
    gemm_kernel<<<dim3(16, 4), 256, 0, stream>>>(lnl, Wv, bv, vb, 512, 256, 256, 1, 0);
    headln_kernel<<<1024, 256, 0, stream>>>(qb, kb);
    pairprep_kernel<<<dim3(512, 8), 256, 0, stream>>>(pair, Wb, Wp, logits, pairv);
    scores_kernel<<<dim3(32, 8, 8), 128, 0, stream>>>(qb, kb, logits);
    softmax_kernel<<<dim3(512, 8), 256, 0, stream>>>(logits, mask);
    attnv_kernel<<<64, 256, 0, stream>>>(logits, vb, attcat);
    attnpair_kernel<<<512, 256, 0, stream>>>(logits, pairv, attcat);
    gemm_kernel<<<dim3(16, 4), 256, 0, stream>>>(attcat, Wo, nullptr, local, 512, 256, 512, 0, 1);
    // --- Update(local) ---
    ln256_kernel<<<512, 256, 0, stream>>>(local, xb, l1s, l1b);
    gemm_kernel<<<dim3(16, 8), 256, 0, stream>>>(xb, U1g, nullptr, gbuf, 512, 512, 256, 0, 0);
    gemm_kernel<<<dim3(16, 8), 256, 0, stream>>>(xb, U1d, nullptr, dbuf, 512, 512, 256, 0, 0);
    gelugate_kernel<<<1024, 256, 0, stream>>>(gbuf, dbuf, hbuf, 262144);
    gemm_kernel<<<dim3(16, 4), 256, 0, stream>>>(hbuf, U1o, nullptr, local, 512, 256, 512, 0, 1);
    // --- LocalToPair ---
    ln256_kernel<<<512, 256, 0, stream>>>(local, lnl, nullptr, nullptr);
    gemm_kernel<<<dim3(16, 2), 256, 0, stream>>>(lnl, Lgl, nullptr, gl, 512, 128, 256, 0, 0);
    gemm_kernel<<<dim3(16, 2), 256, 0, stream>>>(lnl, Lvl, nullptr, vl, 512, 128, 256, 0, 0);
    gemm_kernel<<<dim3(16, 2), 256, 0, stream>>>(lnl, Lgr, nullptr, gr, 512, 128, 256, 0, 0);
    gemm_kernel<<<dim3(16, 2), 256, 0, stream>>>(lnl, Lvr, nullptr, vrb, 512, 128, 256, 0, 0);
    (void)hipMemsetAsync(left, 0, 65536 * sizeof(float), stream);
    (void)hipMemsetAsync(right, 0, 65536 * sizeof(float), stream);
    l2p_kernel<<<dim3(32, 64), 256, 0, stream>>>(pair, Pg, Pv, gl, vl, gr, vrb, mask, left, right);
    // --- pair += concat @ Wl2p ; Update(pair) ---
    pairout_kernel<<<dim3(512, 16), 256, 0, stream>>>(pair, left, right, Wl2p, l2s, l2b,
                                                      U2g, U2d, U2o);
  }
  // final LayerNorms (in place in d_out)
  ln256_kernel<<<512, 256, 0, stream>>>(local, local, F(29), F(30));
  lnpair_kernel<<<32768, 256, 0, stream>>>(pair, F(31), F(32));
}